// TriplaneDecoder_43542378447413
// MI455X (gfx1250) — compile-verified
//
#include <hip/hip_runtime.h>

typedef __attribute__((ext_vector_type(2))) float v2f;
typedef __attribute__((ext_vector_type(8))) float v8f;

#define NPB (128*128*32)   // points per batch (524288)
#define HW  65536          // 256*256

__device__ __forceinline__ v8f wmma_f32_k4(v2f a, v2f b, v8f c) {
  // D(16x16,f32) = A(16x4,f32) * B(4x16,f32) + C
  return __builtin_amdgcn_wmma_f32_16x16x4_f32(false, a, false, b, (short)0, c, false, false);
}

__device__ __forceinline__ float leaky(float v) { return v >= 0.f ? v : 0.01f * v; }

__global__ __launch_bounds__(128)
void triplane_decoder_kernel(const float* __restrict__ tri,
                             const float* __restrict__ pts,
                             const float* __restrict__ W0, const float* __restrict__ b0,
                             const float* __restrict__ W1, const float* __restrict__ b1,
                             const float* __restrict__ W2, const float* __restrict__ b2,
                             float* __restrict__ out)
{
  __shared__ float sW0[32*64];
  __shared__ float sW1[64*64];
  __shared__ float sW2[64*48];      // cols 33..47 zero-padded
  __shared__ float sb0[64];
  __shared__ float sb1[64];
  __shared__ float sb2[48];
  __shared__ float sFeat[4][16*33]; // per-wave 16 points x 32 chans (stride 33)
  __shared__ float sHid [4][16*65]; // per-wave 16 points x 64 hidden (stride 65)

  const int tid = threadIdx.x;

  // ---- stage weights/biases into LDS ----
  for (int i = tid; i < 32*64; i += 128) sW0[i] = W0[i];
  for (int i = tid; i < 64*64; i += 128) sW1[i] = W1[i];
  for (int i = tid; i < 64*48; i += 128) {
    int r = i / 48, c = i % 48;
    sW2[i] = (c < 33) ? W2[r*33 + c] : 0.f;
  }
  if (tid < 64) { sb0[tid] = b0[tid]; sb1[tid] = b1[tid]; }
  if (tid < 48) sb2[tid] = (tid < 33) ? b2[tid] : 0.f;
  __syncthreads();

  const int wave = tid >> 5;
  const int lane = tid & 31;
  const int m    = lane & 15;   // point-in-tile / N index
  const int half = lane >> 4;   // 0: K/ch 0..15 side, 1: 16..31 side

  const long tile = (long)blockIdx.x * 4 + wave;  // 16-point tile
  const long g0   = tile * 16;
  const long gp   = g0 + m;
  const int  b    = (int)(gp / NPB);              // tile never straddles batches

  const float px = pts[gp*3 + 0];
  const float py = pts[gp*3 + 1];
  const float pz = pts[gp*3 + 2];

  float* feat = sFeat[wave];
  float* hid  = sHid[wave];

  // ---- triplane bilinear gather: lane covers (point m, channels half*16..+15) ----
  float acc[16];
  #pragma unroll
  for (int c = 0; c < 16; ++c) acc[c] = 0.f;

  const float* planeBase = tri + (long)b * 3 * 32 * HW + (long)half * 16 * HW;

  #pragma unroll
  for (int pl = 0; pl < 3; ++pl) {
    const float gx = (pl == 2) ? py : px;   // planes: (x,y), (x,z), (y,z)
    const float gy = (pl == 0) ? py : pz;
    const float x = (gx + 1.f) * 128.f - 0.5f;
    const float y = (gy + 1.f) * 128.f - 0.5f;
    const float fx0 = floorf(x), fy0 = floorf(y);
    const float wx = x - fx0,    wy = y - fy0;
    const int x0 = (int)fx0, y0 = (int)fy0;
    const float* pb = planeBase + (long)pl * 32 * HW;
    #pragma unroll
    for (int cy = 0; cy < 2; ++cy) {
      const int  yi  = y0 + cy;
      const bool vy  = (yi >= 0) & (yi < 256);
      const int  yc  = min(max(yi, 0), 255);
      const float wyv = cy ? wy : (1.f - wy);
      #pragma unroll
      for (int cx = 0; cx < 2; ++cx) {
        const int  xi = x0 + cx;
        const bool vx = (xi >= 0) & (xi < 256);
        const int  xc = min(max(xi, 0), 255);
        const float w = ((vx & vy) ? 1.f : 0.f) * wyv * (cx ? wx : (1.f - wx));
        const float* cp = pb + yc * 256 + xc;
        #pragma unroll
        for (int c = 0; c < 16; ++c)           // 16 loads, shared base, imm offsets
          acc[c] = fmaf(w, cp[(long)c * HW], acc[c]);
      }
    }
  }
  #pragma unroll
  for (int c = 0; c < 16; ++c) feat[m*33 + half*16 + c] = acc[c];
  __syncthreads();

  // ---- layer 0: (16x32) @ (32x64) + b0, leaky ----
  v8f d0[4];
  #pragma unroll
  for (int nt = 0; nt < 4; ++nt) {
    const float bv = sb0[nt*16 + m];
    v8f c; 
    #pragma unroll
    for (int r = 0; r < 8; ++r) c[r] = bv;
    #pragma unroll
    for (int ks = 0; ks < 8; ++ks) {
      const int k = ks*4 + 2*half;
      v2f a, w;
      a.x = feat[m*33 + k];
      a.y = feat[m*33 + k + 1];
      w.x = sW0[(k    )*64 + nt*16 + m];
      w.y = sW0[(k + 1)*64 + nt*16 + m];
      c = wmma_f32_k4(a, w, c);
    }
    d0[nt] = c;
  }
  __syncthreads();
  #pragma unroll
  for (int nt = 0; nt < 4; ++nt)
    #pragma unroll
    for (int r = 0; r < 8; ++r)
      hid[(r + 8*half)*65 + nt*16 + m] = leaky(d0[nt][r]);
  __syncthreads();

  // ---- layer 1: (16x64) @ (64x64) + b1, leaky ----
  v8f d1[4];
  #pragma unroll
  for (int nt = 0; nt < 4; ++nt) {
    const float bv = sb1[nt*16 + m];
    v8f c;
    #pragma unroll
    for (int r = 0; r < 8; ++r) c[r] = bv;
    #pragma unroll
    for (int ks = 0; ks < 16; ++ks) {
      const int k = ks*4 + 2*half;
      v2f a, w;
      a.x = hid[m*65 + k];
      a.y = hid[m*65 + k + 1];
      w.x = sW1[(k    )*64 + nt*16 + m];
      w.y = sW1[(k + 1)*64 + nt*16 + m];
      c = wmma_f32_k4(a, w, c);
    }
    d1[nt] = c;
  }
  __syncthreads();
  #pragma unroll
  for (int nt = 0; nt < 4; ++nt)
    #pragma unroll
    for (int r = 0; r < 8; ++r)
      hid[(r + 8*half)*65 + nt*16 + m] = leaky(d1[nt][r]);
  __syncthreads();

  // ---- layer 2: (16x64) @ (64x33) + b2 -> global ----
  #pragma unroll
  for (int nt = 0; nt < 3; ++nt) {
    const float bv = sb2[nt*16 + m];
    v8f c;
    #pragma unroll
    for (int r = 0; r < 8; ++r) c[r] = bv;
    #pragma unroll
    for (int ks = 0; ks < 16; ++ks) {
      const int k = ks*4 + 2*half;
      v2f a, w;
      a.x = hid[m*65 + k];
      a.y = hid[m*65 + k + 1];
      w.x = sW2[(k    )*48 + nt*16 + m];
      w.y = sW2[(k + 1)*48 + nt*16 + m];
      c = wmma_f32_k4(a, w, c);
    }
    const int n = nt*16 + m;
    if (n < 33) {
      #pragma unroll
      for (int r = 0; r < 8; ++r)
        out[(g0 + r + 8*half)*33 + n] = c[r];
    }
  }
}

extern "C" void kernel_launch(void* const* d_in, const int* in_sizes, int n_in,
                              void* d_out, int out_size, void* d_ws, size_t ws_size,
                              hipStream_t stream) {
  const float* tri = (const float*)d_in[0];
  const float* pts = (const float*)d_in[1];
  const float* W0  = (const float*)d_in[2];
  const float* b0  = (const float*)d_in[3];
  const float* W1  = (const float*)d_in[4];
  const float* b1  = (const float*)d_in[5];
  const float* W2  = (const float*)d_in[6];
  const float* b2  = (const float*)d_in[7];
  float* out = (float*)d_out;

  const long total_pts = 2L * NPB;          // 1,048,576
  const int  tiles     = (int)(total_pts / 16); // 65,536 (one wave each)
  const int  blocks    = tiles / 4;         // 4 waves per 128-thread block

  triplane_decoder_kernel<<<blocks, 128, 0, stream>>>(tri, pts, W0, b0, W1, b1, W2, b2, out);
}